// GIN_GCN_Hybrid_19628000543431
// MI455X (gfx1250) — compile-verified
//
#include <hip/hip_runtime.h>
#include <hip/hip_bf16.h>

// ---------------------------------------------------------------------------
// GIN + GCN hybrid GNN for MI455X (gfx1250, wave32).
// Dense node GEMMs -> V_WMMA_F32_16X16X4_F32 (full f32 precision).
// Edge segment-sums -> float4 gather + global_atomic_add_f32 scatter.
// ---------------------------------------------------------------------------

typedef __attribute__((ext_vector_type(2))) float v2f;
typedef __attribute__((ext_vector_type(8))) float v8f;

#define N_NODES 50000
#define N_EDGES 600000
#define N_GRAPHS 256
#define HID 128

#define CDIV(a, b) (((a) + (b) - 1) / (b))

// ---------------------------------------------------------------------------
// fill: zero a float buffer (workspace is poisoned by the harness)
// ---------------------------------------------------------------------------
__global__ void fill_zero_kernel(float* __restrict__ p, int n) {
  int t = blockIdx.x * blockDim.x + threadIdx.x;
  if (t < n) p[t] = 0.0f;
}

// ---------------------------------------------------------------------------
// GEMM: Y[nrows x 128] = act( (X [+ Xadd]) @ W[K x 128] + bias )
// One wave -> 16 rows x 128 cols. K in {64, 128}.
//
// W is staged into LDS K-pair-interleaved:
//   sW[(k>>1)*256 + 2*c + (k&1)] = W[k][c]
// so a WMMA B-fragment (W[kb][c], W[kb+1][c]) is one aligned ds_load_b64
// landing directly in an even VGPR pair (no repacking movs).
//
// A-fragment layout (f32 16x4): lanes 0-15 hold K=k0..k0+1, lanes 16-31 hold
// K=k0+2..k0+3 for row = lane&15. D layout: lane half selects M or M+8.
// HAS_ADD specialization keeps the K-loop free of exec-mask branches.
// ---------------------------------------------------------------------------
template <int HAS_ADD>
__global__ __launch_bounds__(256) void gemm_wmma_kernel(
    const float* __restrict__ X, const float* __restrict__ Xadd,
    const float* __restrict__ W, const float* __restrict__ bias,
    float* __restrict__ Y, int nrows, int K, int do_relu) {
  __shared__ float sW[128 * HID];  // 64 KB max (K=128)

  const int tid = threadIdx.x;
  // cooperatively stage W (K x 128 row-major) into pair-interleaved LDS
  {
    const int nvec = K * (HID / 4);
    const float4* Wv = (const float4*)W;
    for (int i = tid; i < nvec; i += 256) {
      float4 v = Wv[i];
      const int k = i >> 5;            // row
      const int cb = (i & 31) * 4;     // first column of this float4
      float* dstp = sW + (k >> 1) * 256 + (k & 1);
      dstp[2 * (cb + 0)] = v.x;
      dstp[2 * (cb + 1)] = v.y;
      dstp[2 * (cb + 2)] = v.z;
      dstp[2 * (cb + 3)] = v.w;
    }
  }
  __syncthreads();

  const int wave = tid >> 5;
  const int lane = tid & 31;
  const int rowBase = (blockIdx.x * 8 + wave) * 16;
  if (rowBase >= nrows) return;  // wave-uniform: EXEC stays all-ones

  const int mrow = lane & 15;         // A row within tile
  const int khalf = (lane >> 4) * 2;  // 0 for lanes 0-15, 2 for lanes 16-31
  const int col2 = (lane & 15) * 2;   // interleaved column offset

  v8f acc[8];
#pragma unroll
  for (int nt = 0; nt < 8; ++nt)
    acc[nt] = (v8f){0.f, 0.f, 0.f, 0.f, 0.f, 0.f, 0.f, 0.f};

  const float* xrow = X + (size_t)(rowBase + mrow) * K;
  const float* arow = HAS_ADD ? (Xadd + (size_t)(rowBase + mrow) * K) : nullptr;

#pragma unroll 2
  for (int k0 = 0; k0 < K; k0 += 4) {
    const int kb = k0 + khalf;  // even -> 8B aligned float2 load
    float2 av = *(const float2*)(xrow + kb);
    if (HAS_ADD) {
      float2 ad = *(const float2*)(arow + kb);
      av.x += ad.x;
      av.y += ad.y;
    }
    v2f a;
    a.x = av.x;
    a.y = av.y;
    const float* wrow = sW + (kb >> 1) * 256 + col2;  // fragment base
#pragma unroll
    for (int nt = 0; nt < 8; ++nt) {
      v2f b = *(const v2f*)(wrow + nt * 32);  // one ds_load_b64 per fragment
      acc[nt] = __builtin_amdgcn_wmma_f32_16x16x4_f32(
          /*neg_a=*/false, a, /*neg_b=*/false, b,
          /*c_mod=*/(short)0, acc[nt], /*reuse_a=*/false, /*reuse_b=*/false);
    }
  }

  const int rhalf = (lane >> 4) * 8;  // D: lane half selects M or M+8
  const int col = lane & 15;
#pragma unroll
  for (int nt = 0; nt < 8; ++nt) {
    const int c = nt * 16 + col;
    const float bv = bias ? bias[c] : 0.0f;
#pragma unroll
    for (int v = 0; v < 8; ++v) {
      const int r = rowBase + v + rhalf;
      float val = acc[nt][v] + bv;
      if (do_relu) val = fmaxf(val, 0.0f);
      Y[(size_t)r * HID + c] = val;
    }
  }
}

// ---------------------------------------------------------------------------
// GIN scatter: agg[dst] += h[src]   (D = 64 or 128, 4 floats per thread)
// ---------------------------------------------------------------------------
__global__ void scatter_sum_kernel(const float* __restrict__ h,
                                   const int* __restrict__ src,
                                   const int* __restrict__ dst,
                                   float* __restrict__ agg, int nEdges, int D,
                                   int shift) {
  int t = blockIdx.x * blockDim.x + threadIdx.x;
  int total = nEdges << shift;
  if (t >= total) return;
  int e = t >> shift;
  int g = (t & ((1 << shift) - 1)) * 4;
  int s = src[e], d = dst[e];
  float4 v = *(const float4*)(h + (size_t)s * D + g);
  float* o = agg + (size_t)d * D + g;
  atomicAdd(o + 0, v.x);
  atomicAdd(o + 1, v.y);
  atomicAdd(o + 2, v.z);
  atomicAdd(o + 3, v.w);
}

// ---------------------------------------------------------------------------
// GCN normalized scatter: out[dst] += hw[src] * dinv[src] * dinv[dst]
// ---------------------------------------------------------------------------
__global__ void gcn_scatter_kernel(const float* __restrict__ hw,
                                   const int* __restrict__ src,
                                   const int* __restrict__ dst,
                                   const float* __restrict__ dinv,
                                   float* __restrict__ out, int nEdges) {
  int t = blockIdx.x * blockDim.x + threadIdx.x;
  int total = nEdges << 5;  // D=128 -> 32 groups of 4
  if (t >= total) return;
  int e = t >> 5;
  int g = (t & 31) * 4;
  int s = src[e], d = dst[e];
  float sc = dinv[s] * dinv[d];
  float4 v = *(const float4*)(hw + (size_t)s * HID + g);
  float* o = out + (size_t)d * HID + g;
  atomicAdd(o + 0, v.x * sc);
  atomicAdd(o + 1, v.y * sc);
  atomicAdd(o + 2, v.z * sc);
  atomicAdd(o + 3, v.w * sc);
}

// GCN finalize: h = relu(h + hw * dinv^2 (self loop) + bias)
__global__ void gcn_finalize_kernel(float* __restrict__ h,
                                    const float* __restrict__ hw,
                                    const float* __restrict__ dinv,
                                    const float* __restrict__ bias,
                                    int nNodes) {
  int t = blockIdx.x * blockDim.x + threadIdx.x;
  int total = nNodes * HID;
  if (t >= total) return;
  int i = t >> 7;
  int c = t & 127;
  float di = dinv[i];
  float val = h[t] + hw[t] * di * di + bias[c];
  h[t] = fmaxf(val, 0.0f);
}

// degree accumulate (in-degree) and D^{-1/2} finalize (+1 self loop)
__global__ void deg_kernel(const int* __restrict__ dst, float* __restrict__ deg,
                           int nEdges) {
  int t = blockIdx.x * blockDim.x + threadIdx.x;
  if (t < nEdges) atomicAdd(&deg[dst[t]], 1.0f);
}
__global__ void dinv_kernel(float* __restrict__ deg, int nNodes) {
  int t = blockIdx.x * blockDim.x + threadIdx.x;
  if (t < nNodes) {
    float d = deg[t] + 1.0f;  // self loop
    deg[t] = rsqrtf(fmaxf(d, 1.0f));
  }
}

// ---------------------------------------------------------------------------
// pooling: pooled[batch[i]] += h[i]   (D=128)
// ---------------------------------------------------------------------------
__global__ void pool_kernel(const float* __restrict__ h,
                            const int* __restrict__ batch,
                            float* __restrict__ pooled, int nNodes) {
  int t = blockIdx.x * blockDim.x + threadIdx.x;
  int total = nNodes << 5;
  if (t >= total) return;
  int i = t >> 5;
  int g = (t & 31) * 4;
  int b = batch[i];
  float4 v = *(const float4*)(h + (size_t)i * HID + g);
  float* o = pooled + (size_t)b * HID + g;
  atomicAdd(o + 0, v.x);
  atomicAdd(o + 1, v.y);
  atomicAdd(o + 2, v.z);
  atomicAdd(o + 3, v.w);
}

__global__ void cnt_kernel(const int* __restrict__ batch,
                           float* __restrict__ cnt, int nNodes) {
  int t = blockIdx.x * blockDim.x + threadIdx.x;
  if (t < nNodes) atomicAdd(&cnt[batch[t]], 1.0f);
}

// ---------------------------------------------------------------------------
// final FC: out[g] = concat(pooled1[g], pooled2[g]/cnt[g]) @ fc_w + fc_b
// one block per graph, 128 threads (one per output column)
// ---------------------------------------------------------------------------
__global__ __launch_bounds__(128) void fc_kernel(
    const float* __restrict__ pooled1, const float* __restrict__ pooled2,
    const float* __restrict__ cnt, const float* __restrict__ fcw,
    const float* __restrict__ fcb, float* __restrict__ out) {
  __shared__ float srow[256];
  int g = blockIdx.x;
  int j = threadIdx.x;
  srow[j] = pooled1[g * HID + j];
  float c = fmaxf(cnt[g], 1.0f);
  srow[HID + j] = pooled2[g * HID + j] / c;
  __syncthreads();
  float acc = fcb[j];
#pragma unroll 8
  for (int k = 0; k < 256; ++k) acc += srow[k] * fcw[k * HID + j];
  out[g * HID + j] = acc;
}

// ---------------------------------------------------------------------------
extern "C" void kernel_launch(void* const* d_in, const int* in_sizes, int n_in,
                              void* d_out, int out_size, void* d_ws,
                              size_t ws_size, hipStream_t stream) {
  const int N = N_NODES, E = N_EDGES, G = N_GRAPHS;

  const float* x = (const float*)d_in[0];
  const int* src = (const int*)d_in[1];
  const int* dst = src + E;
  const int* batch = (const int*)d_in[2];
  const float* g_w1[3] = {(const float*)d_in[3], (const float*)d_in[7],
                          (const float*)d_in[11]};
  const float* g_b1[3] = {(const float*)d_in[4], (const float*)d_in[8],
                          (const float*)d_in[12]};
  const float* g_w2[3] = {(const float*)d_in[5], (const float*)d_in[9],
                          (const float*)d_in[13]};
  const float* g_b2[3] = {(const float*)d_in[6], (const float*)d_in[10],
                          (const float*)d_in[14]};
  const float* c_w[3] = {(const float*)d_in[15], (const float*)d_in[17],
                         (const float*)d_in[19]};
  const float* c_b[3] = {(const float*)d_in[16], (const float*)d_in[18],
                         (const float*)d_in[20]};
  const float* fcw = (const float*)d_in[21];
  const float* fcb = (const float*)d_in[22];
  float* out = (float*)d_out;

  // workspace layout
  const int NSZ = N * HID;  // 6,400,000 floats
  float* ws = (float*)d_ws;
  float* bufA = ws;
  float* bufB = bufA + NSZ;
  float* bufC = bufB + NSZ;
  float* dinv = bufC + NSZ;
  float* pooled1 = dinv + N;
  float* pooled2 = pooled1 + G * HID;
  float* cnt = pooled2 + G * HID;

  const int TB = 256;
  const int gemmBlocks = CDIV(N / 16, 8);  // 8 waves/block, 16 rows/wave

#define FILL(p, n) fill_zero_kernel<<<CDIV((n), TB), TB, 0, stream>>>((p), (n))
#define GEMM_ADD(X_, XA_, W_, B_, Y_, K_, RELU_)                          \
  gemm_wmma_kernel<1><<<gemmBlocks, TB, 0, stream>>>((X_), (XA_), (W_),   \
                                                     (B_), (Y_), N, (K_), \
                                                     (RELU_))
#define GEMM(X_, W_, B_, Y_, K_, RELU_)                                   \
  gemm_wmma_kernel<0><<<gemmBlocks, TB, 0, stream>>>((X_), nullptr, (W_), \
                                                     (B_), (Y_), N, (K_), \
                                                     (RELU_))

  // ================= GIN branch =================
  // layer 0 (K = 64)
  FILL(bufA, N * 64);
  scatter_sum_kernel<<<CDIV(E * 16, TB), TB, 0, stream>>>(x, src, dst, bufA, E,
                                                          64, 4);
  GEMM_ADD(x, bufA, g_w1[0], g_b1[0], bufB, 64, 1);
  GEMM(bufB, g_w2[0], g_b2[0], bufC, HID, 1);
  // layer 1 (h in bufC)
  FILL(bufA, NSZ);
  scatter_sum_kernel<<<CDIV(E * 32, TB), TB, 0, stream>>>(bufC, src, dst, bufA,
                                                          E, HID, 5);
  GEMM_ADD(bufC, bufA, g_w1[1], g_b1[1], bufB, HID, 1);
  GEMM(bufB, g_w2[1], g_b2[1], bufA, HID, 1);
  // layer 2 (h in bufA)
  FILL(bufC, NSZ);
  scatter_sum_kernel<<<CDIV(E * 32, TB), TB, 0, stream>>>(bufA, src, dst, bufC,
                                                          E, HID, 5);
  GEMM_ADD(bufA, bufC, g_w1[2], g_b1[2], bufB, HID, 1);
  GEMM(bufB, g_w2[2], g_b2[2], bufC, HID, 1);
  // global add pool (h in bufC)
  FILL(pooled1, G * HID);
  pool_kernel<<<CDIV(N * 32, TB), TB, 0, stream>>>(bufC, batch, pooled1, N);

  // ================= GCN branch =================
  FILL(dinv, N);
  deg_kernel<<<CDIV(E, TB), TB, 0, stream>>>(dst, dinv, E);
  dinv_kernel<<<CDIV(N, TB), TB, 0, stream>>>(dinv, N);

  // layer 0 (K = 64): hw = x @ w
  GEMM(x, c_w[0], nullptr, bufA, 64, 0);
  FILL(bufB, NSZ);
  gcn_scatter_kernel<<<CDIV(E * 32, TB), TB, 0, stream>>>(bufA, src, dst, dinv,
                                                          bufB, E);
  gcn_finalize_kernel<<<CDIV(NSZ, TB), TB, 0, stream>>>(bufB, bufA, dinv,
                                                        c_b[0], N);
  // layer 1 (h in bufB)
  GEMM(bufB, c_w[1], nullptr, bufA, HID, 0);
  FILL(bufC, NSZ);
  gcn_scatter_kernel<<<CDIV(E * 32, TB), TB, 0, stream>>>(bufA, src, dst, dinv,
                                                          bufC, E);
  gcn_finalize_kernel<<<CDIV(NSZ, TB), TB, 0, stream>>>(bufC, bufA, dinv,
                                                        c_b[1], N);
  // layer 2 (h in bufC)
  GEMM(bufC, c_w[2], nullptr, bufA, HID, 0);
  FILL(bufB, NSZ);
  gcn_scatter_kernel<<<CDIV(E * 32, TB), TB, 0, stream>>>(bufA, src, dst, dinv,
                                                          bufB, E);
  gcn_finalize_kernel<<<CDIV(NSZ, TB), TB, 0, stream>>>(bufB, bufA, dinv,
                                                        c_b[2], N);
  // global mean pool (h in bufB)
  FILL(pooled2, G * HID);
  FILL(cnt, G);
  pool_kernel<<<CDIV(N * 32, TB), TB, 0, stream>>>(bufB, batch, pooled2, N);
  cnt_kernel<<<CDIV(N, TB), TB, 0, stream>>>(batch, cnt, N);

  // ================= final FC =================
  fc_kernel<<<G, HID, 0, stream>>>(pooled1, pooled2, cnt, fcw, fcb, out);

#undef FILL
#undef GEMM
#undef GEMM_ADD
}